// GCN_31851477467844
// MI455X (gfx1250) — compile-verified
//
#include <hip/hip_runtime.h>
#include <hip/hip_bf16.h>

// ---------------------------------------------------------------------------
// GCN (4x GCNConv + BN + ReLU, JK-cat, segment pool, 2-layer MLP) for MI455X.
// fp32 GEMMs run on the CDNA5 matrix pipe via V_WMMA_F32_16X16X4_F32 (native
// fp32 WMMA -> bit-compatible precision with the fp32 reference). Edge
// aggregation is scatter-atomic into an L2-resident feature buffer (25.6 MB
// << 192 MB L2), with per-edge norms precomputed once.
// ---------------------------------------------------------------------------

typedef float v2f __attribute__((ext_vector_type(2)));
typedef float v8f __attribute__((ext_vector_type(8)));

#define DIMF 64
#define CATF 256
#define OUTF 10

// ------------------------------ utility ------------------------------------
__global__ void zero_f32(float* __restrict__ p, int n) {
    int i = blockIdx.x * blockDim.x + threadIdx.x;
    if (i < n) p[i] = 0.0f;
}

// deg[dst] += 1 over all edges
__global__ void deg_kernel(const int* __restrict__ dst, float* __restrict__ deg, int E) {
    int e = blockIdx.x * blockDim.x + threadIdx.x;
    if (e < E) atomicAdd(&deg[dst[e]], 1.0f);
}

// dinv = rsqrt(deg + 1)   (self-loop)
__global__ void dinv_kernel(float* __restrict__ deg, int N) {
    int i = blockIdx.x * blockDim.x + threadIdx.x;
    if (i < N) deg[i] = __frsqrt_rn(deg[i] + 1.0f);
}

// enorm[e] = dinv[src[e]] * dinv[dst[e]]
__global__ void enorm_kernel(const int* __restrict__ src, const int* __restrict__ dst,
                             const float* __restrict__ dinv, float* __restrict__ enorm, int E) {
    int e = blockIdx.x * blockDim.x + threadIdx.x;
    if (e < E) enorm[e] = dinv[src[e]] * dinv[dst[e]];
}

// ------------------------- WMMA fp32 GEMM (h @ W) --------------------------
// out[N,64] = A[N,64] @ W[64,64], row-major. One 256-thread block = 8 waves;
// each wave computes a 16x16 tile; block covers 32 rows x 64 cols.
// N must be a multiple of 32 (100000 = 32 * 3125) so EXEC is all-ones.
__global__ __launch_bounds__(256) void gemm_n64_wmma(const float* __restrict__ A,
                                                     const float* __restrict__ W,
                                                     float* __restrict__ out) {
    const int wid  = threadIdx.x >> 5;        // 0..7
    const int lane = threadIdx.x & 31;
    const int m0   = blockIdx.x * 32 + (wid >> 2) * 16;  // row tile
    const int n0   = (wid & 3) * 16;                     // col tile
    const int mr   = lane & 15;               // M row / N col within tile
    const int kh   = (lane >> 4) * 2;         // 0 (lanes 0-15) or 2 (lanes 16-31)

    const float* __restrict__ arow = A + (size_t)(m0 + mr) * DIMF;

    v8f c = {};
#pragma unroll
    for (int k = 0; k < DIMF; k += 4) {
        v2f a, b;
        // A 16x4 fp32 fragment: VGPR0 = K=0 / K=2, VGPR1 = K=1 / K=3
        a.x = arow[k + kh];
        a.y = arow[k + kh + 1];
        // B 4x16 fp32 fragment: same K split, N striped across lanes
        b.x = W[(k + kh) * DIMF + n0 + mr];
        b.y = W[(k + kh + 1) * DIMF + n0 + mr];
        c = __builtin_amdgcn_wmma_f32_16x16x4_f32(
                /*neg_a=*/false, a, /*neg_b=*/false, b,
                /*c_mod=*/(short)0, c, /*reuse_a=*/false, /*reuse_b=*/false);
    }

    // C/D layout: VGPR r -> (M = r, N = lane) lanes 0-15 ; (M = r+8) lanes 16-31
    const int col   = n0 + mr;
    const int rbase = m0 + (lane >> 4) * 8;
#pragma unroll
    for (int r = 0; r < 8; ++r)
        out[(size_t)(rbase + r) * DIMF + col] = c[r];
}

// h_next[i,d] = h_lin[i,d] * dinv[i]^2 + bias[d]   (self-loop term + bias)
__global__ void agg_init_kernel(const float* __restrict__ h_lin,
                                const float* __restrict__ dinv,
                                const float* __restrict__ bias,
                                float* __restrict__ h_next, int N) {
    int idx = blockIdx.x * blockDim.x + threadIdx.x;
    if (idx >= N * DIMF) return;
    int i = idx >> 6, d = idx & 63;
    float di = dinv[i];
    h_next[idx] = h_lin[idx] * di * di + bias[d];
}

// Edge scatter: h_next[dst] += h_lin[src] * enorm[e]. 16 threads/edge, each
// handles 4 contiguous dims via one float4 load + 4 f32 global atomics (L2).
__global__ void scatter_kernel(const int* __restrict__ src, const int* __restrict__ dst,
                               const float* __restrict__ enorm,
                               const float* __restrict__ h_lin,
                               float* __restrict__ h_next, int E) {
    long long idx = (long long)blockIdx.x * blockDim.x + threadIdx.x;
    if (idx >= (long long)E * 16) return;
    int e  = (int)(idx >> 4);
    int d0 = ((int)idx & 15) * 4;
    float w = enorm[e];
    const float4 v = *(const float4*)(h_lin + (size_t)src[e] * DIMF + d0);
    float* o = h_next + (size_t)dst[e] * DIMF + d0;
    atomicAdd(o + 0, v.x * w);
    atomicAdd(o + 1, v.y * w);
    atomicAdd(o + 2, v.z * w);
    atomicAdd(o + 3, v.w * w);
}

// Fused BN(eval) + ReLU + write next-layer input + pool accumulate.
__global__ void post_kernel(const float* __restrict__ h_next,
                            const float* __restrict__ gamma, const float* __restrict__ beta,
                            const float* __restrict__ mean,  const float* __restrict__ var,
                            const int* __restrict__ batch,
                            float* __restrict__ h_cur, float* __restrict__ pooled,
                            int N, int layer) {
    int idx = blockIdx.x * blockDim.x + threadIdx.x;
    if (idx >= N * DIMF) return;
    int i = idx >> 6, d = idx & 63;
    int ld = layer * DIMF + d;
    float v = (h_next[idx] - mean[ld]) * __frsqrt_rn(var[ld] + 1e-5f) * gamma[ld] + beta[ld];
    v = fmaxf(v, 0.0f);
    h_cur[idx] = v;
    atomicAdd(&pooled[(size_t)batch[i] * CATF + ld], v);
}

// Per-graph MLP: relu(pooled @ lin1_W + b1) @ lin2_W + b2. One 64-thread
// block per graph; pooled row staged in LDS.
__global__ __launch_bounds__(64) void mlp_kernel(const float* __restrict__ pooled,
                                                 const float* __restrict__ W1, const float* __restrict__ b1,
                                                 const float* __restrict__ W2, const float* __restrict__ b2,
                                                 float* __restrict__ out) {
    __shared__ float sp[CATF];
    __shared__ float sh[DIMF];
    int g = blockIdx.x, t = threadIdx.x;
    const float* prow = pooled + (size_t)g * CATF;
#pragma unroll
    for (int j = 0; j < 4; ++j) sp[t + j * DIMF] = prow[t + j * DIMF];
    __syncthreads();
    float acc = b1[t];
#pragma unroll 8
    for (int k = 0; k < CATF; ++k) acc = fmaf(sp[k], W1[k * DIMF + t], acc);
    sh[t] = fmaxf(acc, 0.0f);
    __syncthreads();
    if (t < OUTF) {
        float o = b2[t];
#pragma unroll 8
        for (int d = 0; d < DIMF; ++d) o = fmaf(sh[d], W2[d * OUTF + t], o);
        out[(size_t)g * OUTF + t] = o;
    }
}

// ------------------------------- launcher ----------------------------------
extern "C" void kernel_launch(void* const* d_in, const int* in_sizes, int n_in,
                              void* d_out, int out_size, void* d_ws, size_t ws_size,
                              hipStream_t stream) {
    const float* x    = (const float*)d_in[0];
    const int*   ei   = (const int*)d_in[1];
    const int*   bat  = (const int*)d_in[2];
    const float* W[4] = {(const float*)d_in[4], (const float*)d_in[6],
                         (const float*)d_in[8], (const float*)d_in[10]};
    const float* B[4] = {(const float*)d_in[5], (const float*)d_in[7],
                         (const float*)d_in[9], (const float*)d_in[11]};
    const float* bn_g = (const float*)d_in[12];
    const float* bn_b = (const float*)d_in[13];
    const float* bn_m = (const float*)d_in[14];
    const float* bn_v = (const float*)d_in[15];
    const float* l1W  = (const float*)d_in[16];
    const float* l1b  = (const float*)d_in[17];
    const float* l2W  = (const float*)d_in[18];
    const float* l2b  = (const float*)d_in[19];

    const int N = in_sizes[0] / DIMF;        // 100000
    const int E = in_sizes[1] / 2;           // 1600000
    const int G = out_size / OUTF;           // 1000
    const int* src = ei;
    const int* dst = ei + E;

    // workspace layout (floats); all offsets multiples of 4 for float4 loads
    float* ws     = (float*)d_ws;
    float* dinv   = ws;                              // N
    float* enorm  = dinv + N;                        // E
    float* h_cur  = enorm + E;                       // N*64
    float* h_lin  = h_cur + (size_t)N * DIMF;        // N*64
    float* h_next = h_lin + (size_t)N * DIMF;        // N*64
    float* pooled = h_next + (size_t)N * DIMF;       // G*256

    const int T = 256;
    // init
    zero_f32<<<(N + T - 1) / T, T, 0, stream>>>(dinv, N);
    zero_f32<<<(G * CATF + T - 1) / T, T, 0, stream>>>(pooled, G * CATF);
    // symmetric normalization (computed once, reused by all 4 layers)
    deg_kernel<<<(E + T - 1) / T, T, 0, stream>>>(dst, dinv, E);
    dinv_kernel<<<(N + T - 1) / T, T, 0, stream>>>(dinv, N);
    enorm_kernel<<<(E + T - 1) / T, T, 0, stream>>>(src, dst, dinv, enorm, E);

    const int gemm_blocks = N / 32;                       // N % 32 == 0
    const int nd_blocks   = (N * DIMF + T - 1) / T;
    const long long sc    = (long long)E * 16;
    const int sc_blocks   = (int)((sc + T - 1) / T);

    const float* hin = x;
    for (int l = 0; l < 4; ++l) {
        gemm_n64_wmma<<<gemm_blocks, T, 0, stream>>>(hin, W[l], h_lin);
        agg_init_kernel<<<nd_blocks, T, 0, stream>>>(h_lin, dinv, B[l], h_next, N);
        scatter_kernel<<<sc_blocks, T, 0, stream>>>(src, dst, enorm, h_lin, h_next, E);
        post_kernel<<<nd_blocks, T, 0, stream>>>(h_next, bn_g, bn_b, bn_m, bn_v,
                                                 bat, h_cur, pooled, N, l);
        hin = h_cur;
    }

    mlp_kernel<<<G, DIMF, 0, stream>>>(pooled, l1W, l1b, l2W, l2b, (float*)d_out);
}